// FeatureSynergyModule_49125835931862
// MI455X (gfx1250) — compile-verified
//
#include <hip/hip_runtime.h>
#include <math.h>

typedef __attribute__((ext_vector_type(2))) float v2f;
typedef __attribute__((ext_vector_type(8))) float v8f;

#define WAVES 8

// ---------------------------------------------------------------------------
// Kernel 1: per-row pipeline via WMMA f32 16x16x4, tiles of 16 batch rows.
// All fragment loads are unconditional: out-of-range K indices are steered to
// guaranteed-zero pad slots (xs[.][15], ms[.][9..11], wks row 3, padded w1s/w2s),
// so no EXEC save/restore around LDS loads.
// ---------------------------------------------------------------------------
__global__ __launch_bounds__(256) void fsm_main(
    const float* __restrict__ x,
    const float* __restrict__ w_num, const float* __restrict__ b_num,
    const float* __restrict__ oc1_w, const float* __restrict__ oc1_b,
    const float* __restrict__ oc2_w, const float* __restrict__ oc2_b,
    const float* __restrict__ dc_w,  const float* __restrict__ dc_b,
    float* __restrict__ out, float* __restrict__ partial, int B)
{
    // Block-shared zero-padded weights (B-fragment friendly layouts).
    __shared__ float w1s[6][32][8];    // w1s[p][n][k] = oc1_w[p][n][k][1][1]
    __shared__ float w2s[6][32][20];   // w2s[p][n][k] = oc2_w[p][n][k][1][1]
    __shared__ float wks[6][4][28];    // wks[p][o][i*9+kk] = dc_w[p][o][i][kk/3][kk%3]; row 3 = zeros
    __shared__ float b1s[6][32];
    __shared__ float b2s[6][32];
    __shared__ float wnum[12];         // 9 weights + 3 bias
    // Per-wave staging.
    __shared__ float sx[WAVES][16][16];   // [row][0..11]=raw x, [12..14]=num_e, [15]=0
    __shared__ float sh[WAVES][16][20];   // relu(h), cols 18,19 zero
    __shared__ float sm[WAVES][16][12];   // m, cols 0..8 used, 9..11 zero
    __shared__ float sacc[WAVES][16][12]; // combined accumulator
    __shared__ float psum[WAVES][32][3];  // per-lane column partial sums

    const int tid = threadIdx.x;

    for (int i = tid; i < 6*32*8;  i += 256) ((float*)w1s)[i] = 0.f;
    for (int i = tid; i < 6*32*20; i += 256) ((float*)w2s)[i] = 0.f;
    for (int i = tid; i < 6*4*28;  i += 256) ((float*)wks)[i] = 0.f;
    for (int i = tid; i < 6*32;    i += 256) { ((float*)b1s)[i] = 0.f; ((float*)b2s)[i] = 0.f; }
    __syncthreads();
    for (int i = tid; i < 6*18*6; i += 256) {
        int k = i % 6, n = (i / 6) % 18, p = i / (6*18);
        w1s[p][n][k] = oc1_w[((p*18 + n)*6 + k)*9 + 4];
    }
    for (int i = tid; i < 6*18*18; i += 256) {
        int k = i % 18, n = (i / 18) % 18, p = i / (18*18);
        w2s[p][n][k] = oc2_w[((p*18 + n)*18 + k)*9 + 4];
    }
    for (int i = tid; i < 6*3*27; i += 256) {
        int c = i % 27, n = (i / 27) % 3, p = i / (27*3);
        wks[p][n][c] = dc_w[((p*3 + n)*3 + (c/9))*9 + (c % 9)];
    }
    for (int i = tid; i < 6*18; i += 256) {
        b1s[i/18][i%18] = oc1_b[i];
        b2s[i/18][i%18] = oc2_b[i];
    }
    if (tid < 9) wnum[tid] = w_num[tid];
    if (tid < 3) wnum[9 + tid] = b_num[tid];
    __syncthreads();

    const int lane = tid & 31;
    const int wv   = tid >> 5;
    const int half = lane >> 4;     // which 16-lane half
    const int l16  = lane & 15;

    float (*xs)[16] = sx[wv];
    float (*hs)[20] = sh[wv];
    float (*ms)[12] = sm[wv];
    float (*as)[12] = sacc[wv];

    // zero sm pad cols 9..11 once (never written again)
    if (lane < 16) { ms[lane][9] = 0.f; ms[lane][10] = 0.f; ms[lane][11] = 0.f; }

    float acc0 = 0.f, acc1 = 0.f, acc2 = 0.f;   // column partial sums (3 cols per lane)

    const int ntiles  = (B + 15) >> 4;
    const int gw      = blockIdx.x * WAVES + wv;
    const int wstride = gridDim.x * WAVES;
    const int wkrow   = (l16 < 3) ? l16 : 3;    // clamped index into zero-padded wks

    for (int tile = gw; tile < ntiles; tile += wstride) {
        const int row0 = tile * 16;

        // ---- load x tile (16 rows x 12 cols), coalesced ----
        #pragma unroll
        for (int q = 0; q < 6; ++q) {
            int f = lane + 32*q, r = f / 12, c = f % 12, gr = row0 + r;
            xs[r][c] = (gr < B) ? x[gr*12 + c] : 0.f;
        }
        __builtin_amdgcn_wave_barrier();
        // num_e = num @ w_num^T + b_num  -> cols 12..14 ; col 15 = 0 pad
        if (lane < 16) {
            int r = lane;
            float n0 = xs[r][9], n1 = xs[r][10], n2 = xs[r][11];
            #pragma unroll
            for (int j = 0; j < 3; ++j)
                xs[r][12 + j] = wnum[9 + j] + n0*wnum[j*3+0] + n1*wnum[j*3+1] + n2*wnum[j*3+2];
            xs[r][15] = 0.f;
        }
        // init combined accumulator with raw features
        #pragma unroll
        for (int q = 0; q < 6; ++q) {
            int f = lane + 32*q;
            as[f/12][f%12] = xs[f/12][f%12];
        }
        __builtin_amdgcn_wave_barrier();

        #pragma unroll
        for (int p = 0; p < 6; ++p) {
            constexpr int f1b_[6]   = {0, 0, 0, 3, 3, 6};
            constexpr int f2b_[6]   = {3, 6, 12, 6, 12, 12};
            constexpr int gmask_[6] = {0b011, 0b101, 0b001, 0b110, 0b010, 0b100};
            const int f1b = f1b_[p], f2b = f2b_[p], gmask = gmask_[p];

            // comb column index, steering k>=6 to the zero pad col 15
            auto combcol = [&](int k) -> int {
                return (k < 3) ? (f1b + k) : ((k < 6) ? (f2b + k - 3) : 15);
            };

            // ---- layer1: h = relu(comb @ w1^T + b1)   (2 N-tiles x 2 K-blocks) ----
            #pragma unroll
            for (int u = 0; u < 2; ++u) {
                const int n = l16 + 16*u;
                v8f c;
                const float bias = b1s[p][n];
                #pragma unroll
                for (int r = 0; r < 8; ++r) c[r] = bias;
                #pragma unroll
                for (int t = 0; t < 2; ++t) {
                    const int kb = t*4 + half*2;
                    v2f a, b;
                    a[0] = xs[l16][combcol(kb)];
                    a[1] = xs[l16][combcol(kb + 1)];
                    b[0] = w1s[p][n][kb]; b[1] = w1s[p][n][kb+1];
                    c = __builtin_amdgcn_wmma_f32_16x16x4_f32(false, a, false, b, (short)0, c, false, false);
                }
                if (u == 0) {
                    #pragma unroll
                    for (int r = 0; r < 8; ++r) {
                        float v = c[r]; v = v > 0.f ? v : 0.f;
                        hs[r + 8*half][n] = v;
                    }
                } else if (l16 < 2) {
                    #pragma unroll
                    for (int r = 0; r < 8; ++r) {
                        float v = c[r]; v = v > 0.f ? v : 0.f;
                        hs[r + 8*half][n] = v;
                    }
                } else if (l16 < 4) {       // zero K-pad cols 18,19
                    #pragma unroll
                    for (int r = 0; r < 8; ++r) hs[r + 8*half][n] = 0.f;
                }
            }
            __builtin_amdgcn_wave_barrier();

            // ---- layer2: off = h @ w2^T + b2   (2 N-tiles x 5 K-blocks) ----
            v8f off0, off1;
            {
                const float bi0 = b2s[p][l16], bi1 = b2s[p][l16 + 16];
                #pragma unroll
                for (int r = 0; r < 8; ++r) { off0[r] = bi0; off1[r] = bi1; }
            }
            #pragma unroll
            for (int t = 0; t < 5; ++t) {
                const int kb = t*4 + half*2;
                v2f a, b;
                a[0] = hs[l16][kb]; a[1] = hs[l16][kb+1];
                b[0] = w2s[p][l16][kb]; b[1] = w2s[p][l16][kb+1];
                off0 = __builtin_amdgcn_wmma_f32_16x16x4_f32(false, a, false, b, (short)0, off0, false, false);
                b[0] = w2s[p][l16+16][kb]; b[1] = w2s[p][l16+16][kb+1];
                off1 = __builtin_amdgcn_wmma_f32_16x16x4_f32(false, a, false, b, (short)0, off1, false, false);
            }

            // ---- m[j] = relu(1-|ky+offy|)*relu(1-|kx+offx|), pairs via shfl ----
            {
                const int j = l16 >> 1;                         // j = 0..7 for tile 0
                const float ky = (float)(j / 3) - 1.f;
                const float kx = (float)(j % 3) - 1.f;
                #pragma unroll
                for (int r = 0; r < 8; ++r) {
                    float v = off0[r];
                    float w = __shfl_xor(v, 1, 32);
                    if ((l16 & 1) == 0) {
                        float yy = ky + v, xx = kx + w;
                        float m1 = 1.f - fabsf(yy); m1 = m1 > 0.f ? m1 : 0.f;
                        float m2 = 1.f - fabsf(xx); m2 = m2 > 0.f ? m2 : 0.f;
                        ms[r + 8*half][j] = m1 * m2;
                    }
                }
                #pragma unroll
                for (int r = 0; r < 8; ++r) {                   // j = 8 (cols 16,17)
                    float v = off1[r];
                    float w = __shfl_xor(v, 1, 32);
                    if (l16 == 0) {
                        float yy = 1.f + v, xx = 1.f + w;       // ky=kx=1 for j=8
                        float m1 = 1.f - fabsf(yy); m1 = m1 > 0.f ? m1 : 0.f;
                        float m2 = 1.f - fabsf(xx); m2 = m2 > 0.f ? m2 : 0.f;
                        ms[r + 8*half][8] = m1 * m2;
                    }
                }
            }
            __builtin_amdgcn_wave_barrier();

            // ---- aligned = (f1 x m) @ wk^T + dc_b   (1 N-tile x 7 K-blocks, K=27) ----
            {
                v8f al;
                const float bias = (l16 < 3) ? dc_b[p*3 + l16] : 0.f;
                #pragma unroll
                for (int r = 0; r < 8; ++r) al[r] = bias;
                #pragma unroll
                for (int t = 0; t < 7; ++t) {
                    const int k0 = t*4 + half*2;
                    const int k1 = k0 + 1;
                    // steer k>=27 to zero pad: ms col 9 is always 0
                    const int i0 = (k0 < 27) ? (k0 / 9) : 2;
                    const int m0 = (k0 < 27) ? (k0 % 9) : 9;
                    const int i1 = (k1 < 27) ? (k1 / 9) : 2;
                    const int m1 = (k1 < 27) ? (k1 % 9) : 9;
                    v2f a, b;
                    a[0] = xs[l16][f1b + i0] * ms[l16][m0];
                    a[1] = xs[l16][f1b + i1] * ms[l16][m1];
                    b[0] = wks[p][wkrow][k0];
                    b[1] = wks[p][wkrow][k1];
                    al = __builtin_amdgcn_wmma_f32_16x16x4_f32(false, a, false, b, (short)0, al, false, false);
                }
                if (l16 < 3) {
                    #pragma unroll
                    for (int g = 0; g < 3; ++g) {
                        if (gmask & (1 << g)) {
                            #pragma unroll
                            for (int r = 0; r < 8; ++r)
                                as[r + 8*half][g*3 + l16] += al[r];
                        }
                    }
                }
            }
            __builtin_amdgcn_wave_barrier();
        } // phases

        // ---- write combined, accumulate column sums ----
        #pragma unroll
        for (int q = 0; q < 6; ++q) {
            int f = lane + 32*q, r = f / 12, c = f % 12, gr = row0 + r;
            if (gr < B) {
                float v = as[r][c];
                if (c < 9) v *= 0.25f;
                out[gr*12 + c] = v;
                // lane's column for q is (lane%12 + {0,8,4,0,8,4}[q]) % 12 -> slot {0,2,1,0,2,1}
                const int slot = (q == 0 || q == 3) ? 0 : ((q == 2 || q == 5) ? 1 : 2);
                if (slot == 0) acc0 += v; else if (slot == 1) acc1 += v; else acc2 += v;
            }
        }
    } // tiles

    // ---- deterministic fixed-order block reduction of column sums ----
    psum[wv][lane][0] = acc0;
    psum[wv][lane][1] = acc1;
    psum[wv][lane][2] = acc2;
    __syncthreads();
    if (tid < 12) {
        const int c = tid;
        float s = 0.f;
        for (int w = 0; w < WAVES; ++w)
            for (int l = 0; l < 32; ++l) {
                int d = c - (l % 12); d = ((d % 12) + 12) % 12;   // offset in {0..11}
                if ((d & 3) == 0)                                  // lane contributes to c
                    s += psum[w][l][d >> 2];                       // slot: 0->+0, 1->+4, 2->+8
            }
        partial[blockIdx.x * 12 + c] = s;
    }
}

// ---------------------------------------------------------------------------
// Kernel 2: reduce block partials -> mean -> gated MLP -> sigmoid gate (12)
// ---------------------------------------------------------------------------
__global__ void fsm_reduce(const float* __restrict__ partial, int nblocks, int B,
                           const float* __restrict__ fc1_w, const float* __restrict__ fc1_b,
                           const float* __restrict__ fc2_w, const float* __restrict__ fc2_b,
                           float* __restrict__ gate)
{
    __shared__ float g[12];
    const int c = threadIdx.x;
    if (c < 12) {
        float s = 0.f;
        for (int b = 0; b < nblocks; ++b) s += partial[b*12 + c];
        g[c] = s / (float)B;
    }
    __syncthreads();
    if (c == 0) {
        float a[3];
        for (int j = 0; j < 3; ++j) {
            float s = fc1_b[j];
            for (int k = 0; k < 12; ++k) s += g[k] * fc1_w[j*12 + k];
            a[j] = s > 0.f ? s : 0.f;
        }
        for (int o = 0; o < 12; ++o) {
            float s = fc2_b[o];
            for (int j = 0; j < 3; ++j) s += a[j] * fc2_w[o*3 + j];
            gate[o] = 1.f / (1.f + expf(-s));
        }
    }
}

// ---------------------------------------------------------------------------
// Kernel 3: out[row][c] *= gate[c]  (float4-vectorized, in place)
// ---------------------------------------------------------------------------
__global__ __launch_bounds__(256) void fsm_scale(float* __restrict__ out,
                                                 const float* __restrict__ gate, int B)
{
    __shared__ float g[12];
    if (threadIdx.x < 12) g[threadIdx.x] = gate[threadIdx.x];
    __syncthreads();
    const float4 g0 = make_float4(g[0], g[1], g[2],  g[3]);
    const float4 g1 = make_float4(g[4], g[5], g[6],  g[7]);
    const float4 g2 = make_float4(g[8], g[9], g[10], g[11]);
    float4* o4 = (float4*)out;
    const int idx = blockIdx.x * blockDim.x + threadIdx.x;
    const int stride = gridDim.x * blockDim.x;
    for (int r = idx; r < B; r += stride) {
        float4 a = o4[r*3 + 0], b = o4[r*3 + 1], c = o4[r*3 + 2];
        a.x *= g0.x; a.y *= g0.y; a.z *= g0.z; a.w *= g0.w;
        b.x *= g1.x; b.y *= g1.y; b.z *= g1.z; b.w *= g1.w;
        c.x *= g2.x; c.y *= g2.y; c.z *= g2.z; c.w *= g2.w;
        o4[r*3 + 0] = a; o4[r*3 + 1] = b; o4[r*3 + 2] = c;
    }
}

// ---------------------------------------------------------------------------
extern "C" void kernel_launch(void* const* d_in, const int* in_sizes, int n_in,
                              void* d_out, int out_size, void* d_ws, size_t ws_size,
                              hipStream_t stream)
{
    const float* x      = (const float*)d_in[0];
    const float* w_num  = (const float*)d_in[1];
    const float* b_num  = (const float*)d_in[2];
    const float* oc1_w  = (const float*)d_in[3];
    const float* oc1_b  = (const float*)d_in[4];
    const float* oc2_w  = (const float*)d_in[5];
    const float* oc2_b  = (const float*)d_in[6];
    const float* dc_w   = (const float*)d_in[7];
    const float* dc_b   = (const float*)d_in[8];
    const float* fc1_w  = (const float*)d_in[9];
    const float* fc1_b  = (const float*)d_in[10];
    const float* fc2_w  = (const float*)d_in[11];
    const float* fc2_b  = (const float*)d_in[12];
    float* out = (float*)d_out;
    const int B = in_sizes[0] / 12;

    float* gate    = (float*)d_ws;      // [0..11] sigmoid gate
    float* partial = gate + 16;         // [nblocks*12] block partial sums

    int nblocks = 1024;
    const long cap = (long)(ws_size / sizeof(float)) - 16;
    if (cap > 0 && (long)nblocks * 12 > cap) nblocks = (int)(cap / 12);
    if (nblocks < 1) nblocks = 1;
    const int ntiles = (B + 15) >> 4;
    const int needed = (ntiles + WAVES - 1) / WAVES;
    if (nblocks > needed) nblocks = needed;

    fsm_main<<<nblocks, 256, 0, stream>>>(x, w_num, b_num, oc1_w, oc1_b,
                                          oc2_w, oc2_b, dc_w, dc_b,
                                          out, partial, B);
    fsm_reduce<<<1, 32, 0, stream>>>(partial, nblocks, B, fc1_w, fc1_b, fc2_w, fc2_b, gate);
    fsm_scale<<<512, 256, 0, stream>>>(out, gate, B);
}